// BoundSimplexNeuron_Alpha_56513179681399
// MI455X (gfx1250) — compile-verified
//
#include <hip/hip_runtime.h>
#include <hip/hip_bf16.h>

#define OUTN 4096
#define INN  4096
#define SPEC 512

typedef __attribute__((ext_vector_type(2))) float v2f;
typedef __attribute__((ext_vector_type(8))) float v8f;
typedef __attribute__((ext_vector_type(4))) int   v4i;

typedef __attribute__((address_space(1))) v4i glb_v4i;   // global int4
typedef __attribute__((address_space(3))) v4i lds_v4i;   // LDS int4

#if __has_builtin(__builtin_amdgcn_global_load_async_to_lds_b128)
#define HAVE_ASYNC_LDS 1
#else
#define HAVE_ASYNC_LDS 0
#endif

#if HAVE_ASYNC_LDS && __has_builtin(__builtin_amdgcn_s_wait_asynccnt)
#define WAIT_ASYNC(n) __builtin_amdgcn_s_wait_asynccnt(n)
#elif HAVE_ASYNC_LDS
#define WAIT_ASYNC(n) asm volatile("s_wait_asynccnt %0" :: "i"(n) : "memory")
#else
#define WAIT_ASYNC(n) do {} while (0)
#endif

// ---------------------------------------------------------------------------
// Kernel 1: per-output-neuron scalar coefficients (all CROWN/simplex case
// logic collapses to per-o scalars since every coeff except the simplex term
// is scalar * weight[o,i], and relu(bias) folds into a rank-1 c[s] term).
// ---------------------------------------------------------------------------
__global__ void coeff_kernel(const float* __restrict__ bias,
                             const float* __restrict__ lb,
                             const float* __restrict__ ub,
                             const float* __restrict__ alpha,
                             float* __restrict__ p1, float* __restrict__ q1,
                             float* __restrict__ p2, float* __restrict__ P,
                             float* __restrict__ Q,  float* __restrict__ rb2)
{
    int o = blockIdx.x * blockDim.x + threadIdx.x;
    if (o >= OUTN) return;
    float b = bias[o], l = lb[o], u = ub[o], a = alpha[o];
    float oma = 1.0f - a;

    // CROWN relaxation slopes
    float lb_r = fminf(l, 0.0f);
    float ub_r = fmaxf(u, 0.0f);
    ub_r = fmaxf(ub_r, lb_r + 1e-8f);
    float ud  = ub_r / (ub_r - lb_r);
    float ldv = (ud > 0.5f) ? 1.0f : 0.0f;
    float upper_b = -lb_r * ud;
    float lowd = (fabsf(l) >= fabsf(u)) ? 1.0f : 0.0f;

    // crown part contributes a*ud (pos) / a*ld (neg) unconditionally
    float p1v = a * ud;
    float q1v = a * ldv;
    float p2v = 0.0f;
    float ubias_u = 0.0f, ubias_l = 0.0f;

    if (u <= 0.0f) {
        // inactive: simplex coeffs zero
    } else if (l >= 0.0f) {
        p1v += oma; q1v += oma;           // coeff = weight for both bounds
        ubias_u = b; ubias_l = b;
    } else {
        q1v += oma * lowd;                // lower coeff = lower_d * weight
        p2v  = oma;                       // upper coeff = relu(w+b)-relu(b)
        ubias_u = fmaxf(b, 0.0f);
        ubias_l = lowd * b;
    }
    p1[o] = p1v;
    q1[o] = q1v;
    p2[o] = p2v;
    P[o]  = oma * ubias_u + upper_b + ud * a * b;  // pos coefficient for ubias
    Q[o]  = oma * ubias_l + ldv * a * b;           // neg coefficient for ubias
    rb2[o] = p2v * fmaxf(b, 0.0f);                 // rank-1 correction c[s]
}

// ---------------------------------------------------------------------------
// Kernel 2: materialize GEMM LHS matrices transposed [O][S]. 32x32 LDS
// transpose tile (33-stride to dodge bank conflicts) so BOTH the last_uA
// read and the A1t/A2t writes are fully coalesced (a raw scatter here would
// cost as much HBM traffic as the whole GEMM).
// ---------------------------------------------------------------------------
__global__ __launch_bounds__(256)
void buildA_kernel(const float* __restrict__ uA,
                   const float* __restrict__ p1,
                   const float* __restrict__ q1,
                   const float* __restrict__ p2,
                   float* __restrict__ A1t,
                   float* __restrict__ A2t)
{
    __shared__ float t1[32][33];
    __shared__ float t2[32][33];
    int tid = threadIdx.x;
    int lx  = tid & 31;       // column within tile
    int g   = tid >> 5;       // row group 0..7
    int o0  = blockIdx.x * 32;
    int s0  = blockIdx.y * 32;

    float c1 = p1[o0 + lx];
    float c2 = q1[o0 + lx];
    float c3 = p2[o0 + lx];
#pragma unroll
    for (int j = 0; j < 4; ++j) {
        int r = g * 4 + j;                                  // s_rel
        float v = uA[(size_t)(s0 + r) * OUTN + o0 + lx];    // coalesced in o
        float pos = fmaxf(v, 0.0f);
        float neg = fminf(v, 0.0f);
        t1[lx][r] = pos * c1 + neg * c2;                    // transposed store
        t2[lx][r] = pos * c3;
    }
    __syncthreads();
#pragma unroll
    for (int j = 0; j < 4; ++j) {
        int r = g * 4 + j;                                  // o_rel
        A1t[(size_t)(o0 + r) * SPEC + s0 + lx] = t1[r][lx]; // coalesced in s
        A2t[(size_t)(o0 + r) * SPEC + s0 + lx] = t2[r][lx];
    }
}

// ---------------------------------------------------------------------------
// Kernel 3: ubias[s] (second output) and c[s] (rank-1 correction).
// ---------------------------------------------------------------------------
__global__ void bias_reduce_kernel(const float* __restrict__ uA,
                                   const float* __restrict__ P,
                                   const float* __restrict__ Q,
                                   const float* __restrict__ rb2,
                                   float* __restrict__ ubias_out,
                                   float* __restrict__ c_out)
{
    int s = blockIdx.x;
    int tid = threadIdx.x;
    float s1 = 0.0f, s2 = 0.0f;
    for (int o = tid; o < OUTN; o += 256) {
        float v = uA[(size_t)s * OUTN + o];
        float pos = fmaxf(v, 0.0f);
        float neg = fminf(v, 0.0f);
        s1 += pos * P[o] + neg * Q[o];
        s2 += pos * rb2[o];
    }
    __shared__ float sh1[256];
    __shared__ float sh2[256];
    sh1[tid] = s1; sh2[tid] = s2;
    __syncthreads();
    for (int st = 128; st > 0; st >>= 1) {
        if (tid < st) { sh1[tid] += sh1[tid + st]; sh2[tid] += sh2[tid + st]; }
        __syncthreads();
    }
    if (tid == 0) { ubias_out[s] = sh1[0]; c_out[s] = sh2[0]; }
}

// ---------------------------------------------------------------------------
// Kernel 4: fused dual GEMM with FP32 WMMA (16x16x4).
//   uA[s,i] = sum_o A1t[o][s]*W[o,i] + A2t[o][s]*relu(W[o,i]+b[o]) - c[s]
// Block = 256 thr (8 waves). Block tile: M=128 (16 rows/wave), N=128.
// Weight streamed 32x128 per step through DOUBLE-BUFFERED LDS using the
// CDNA5 async-to-LDS path (GLOBAL_LOAD_ASYNC_TO_LDS_B128 + s_wait_asynccnt):
// the next tile streams in while WMMAs consume the current one. relu is
// computed in VALU from the shared tile so one weight read feeds both GEMMs.
// ---------------------------------------------------------------------------
__global__ __launch_bounds__(256)
void gemm_kernel(const float* __restrict__ W,
                 const float* __restrict__ bias,
                 const float* __restrict__ A1t,
                 const float* __restrict__ A2t,
                 const float* __restrict__ c,
                 float* __restrict__ out)
{
    __shared__ float wt[2][32 * 128];
    __shared__ float bs[2][32];

    int tid    = threadIdx.x;
    int wave   = tid >> 5;
    int lane   = tid & 31;
    int lane16 = lane & 15;
    int half   = lane >> 4;          // 0: K rows {0,1} / M 0..7 ; 1: K rows {2,3} / M 8..15

    int n0 = blockIdx.x * 128;
    int s0 = blockIdx.y * 128 + wave * 16;
    int srow = s0 + lane16;

    int ldrow = tid >> 5;            // 0..7
    int ldcol = (tid & 31) * 4;      // 0..124

    const v8f vzero = {0.f,0.f,0.f,0.f,0.f,0.f,0.f,0.f};
    v8f acc[8];
#pragma unroll
    for (int i = 0; i < 8; ++i) acc[i] = vzero;

    // stage one 32(K) x 128(N) weight tile into LDS buffer `b`
    auto stage = [&](int oc, int b) {
#pragma unroll
        for (int j = 0; j < 4; ++j) {
            int row = ldrow + j * 8;
            const float* src = &W[(size_t)(oc + row) * INN + n0 + ldcol];
            float* dst = &wt[b][row * 128 + ldcol];
#if HAVE_ASYNC_LDS
            __builtin_amdgcn_global_load_async_to_lds_b128(
                (glb_v4i*)src, (lds_v4i*)dst, 0, 0);
#else
            *reinterpret_cast<float4*>(dst) =
                *reinterpret_cast<const float4*>(src);
#endif
        }
        if (tid < 32) bs[b][tid] = bias[oc + tid];
    };

    stage(0, 0);                               // prologue: tile 0 -> buf 0

    const int NT = OUTN / 32;                  // 128 K-tiles
    for (int t = 0; t < NT; ++t) {
        int b = t & 1;
        if (t + 1 < NT) {
            stage((t + 1) * 32, b ^ 1);        // overlap next fetch w/ compute
            WAIT_ASYNC(4);                     // async in-order => tile t done
        } else {
            WAIT_ASYNC(0);
        }
        __syncthreads();                       // all waves' tile t visible

#pragma unroll
        for (int kk = 0; kk < 8; ++kk) {
            int kRel = kk * 4 + 2 * half;      // K rows for this lane half
            int oA   = t * 32 + kRel;
            // A operands: 16x4 f32, M=lane%16, K striped over VGPR x/y + half
            v2f a1, a2;
            a1.x = A1t[(size_t)oA       * SPEC + srow];
            a1.y = A1t[(size_t)(oA + 1) * SPEC + srow];
            a2.x = A2t[(size_t)oA       * SPEC + srow];
            a2.y = A2t[(size_t)(oA + 1) * SPEC + srow];
            float bx = bs[b][kRel];
            float by = bs[b][kRel + 1];
#pragma unroll
            for (int sub = 0; sub < 8; ++sub) {
                float wx = wt[b][kRel * 128 + sub * 16 + lane16];
                float wy = wt[b][(kRel + 1) * 128 + sub * 16 + lane16];
                v2f bw, br;
                bw.x = wx;                     bw.y = wy;
                br.x = fmaxf(wx + bx, 0.0f);   br.y = fmaxf(wy + by, 0.0f);
                acc[sub] = __builtin_amdgcn_wmma_f32_16x16x4_f32(
                    false, a1, false, bw, (short)0, acc[sub], false, false);
                acc[sub] = __builtin_amdgcn_wmma_f32_16x16x4_f32(
                    false, a2, false, br, (short)0, acc[sub], false, false);
            }
        }
        __syncthreads();                       // readers done before overwrite
    }

    // apply rank-1 correction and store (C layout: VGPR v -> M = v + 8*half)
#pragma unroll
    for (int v = 0; v < 8; ++v) {
        int m = s0 + v + half * 8;
        float cv = c[m];
#pragma unroll
        for (int sub = 0; sub < 8; ++sub) {
            out[(size_t)m * INN + n0 + sub * 16 + lane16] = acc[sub][v] - cv;
        }
    }
}

// ---------------------------------------------------------------------------
extern "C" void kernel_launch(void* const* d_in, const int* in_sizes, int n_in,
                              void* d_out, int out_size, void* d_ws, size_t ws_size,
                              hipStream_t stream) {
    const float* uA    = (const float*)d_in[0];  // (1,512,4096)
    const float* W     = (const float*)d_in[1];  // (4096,4096)
    const float* bias  = (const float*)d_in[2];  // (4096,)
    const float* lb    = (const float*)d_in[3];  // (1,4096)
    const float* ub    = (const float*)d_in[4];  // (1,4096)
    const float* alpha = (const float*)d_in[5];  // (1,4096,1)

    float* out = (float*)d_out;                  // [0,512*4096) uA, then 512 ubias
    float* ws  = (float*)d_ws;

    float* p1  = ws;
    float* q1  = ws + 4096;
    float* p2  = ws + 8192;
    float* P   = ws + 12288;
    float* Q   = ws + 16384;
    float* rb2 = ws + 20480;
    float* c   = ws + 24576;
    float* A1t = ws + 32768;
    float* A2t = A1t + (size_t)OUTN * SPEC;

    coeff_kernel<<<OUTN / 256, 256, 0, stream>>>(bias, lb, ub, alpha,
                                                 p1, q1, p2, P, Q, rb2);
    dim3 tg(OUTN / 32, SPEC / 32);
    buildA_kernel<<<tg, 256, 0, stream>>>(uA, p1, q1, p2, A1t, A2t);
    bias_reduce_kernel<<<SPEC, 256, 0, stream>>>(uA, P, Q, rb2,
                                                 out + (size_t)SPEC * INN, c);
    dim3 grid(INN / 128, SPEC / 128);
    gemm_kernel<<<grid, 256, 0, stream>>>(W, bias, A1t, A2t, c, out);
}